// _ReorderScaledDotProductAttention_53334903881772
// MI455X (gfx1250) — compile-verified
//
#include <hip/hip_runtime.h>

// Problem constants (from the reference): bs=16, heads=16, seq=4096, d=32
#define BSZ 16
#define NH  16
#define SEQ 4096
#define HD  32

#define OUT_OFF   (BSZ * NH * SEQ * HD)   // 33,554,432 floats: start of attn_weights in d_out
#define SCORE_N   (BSZ * NH * HD * HD)    // 262,144 floats per partial-score image
#define SPLIT     8                       // L-split for kv_gemm (ws use = SPLIT MB)

typedef __attribute__((ext_vector_type(2))) float v2f;
typedef __attribute__((ext_vector_type(8))) float v8f;

// ---------------------------------------------------------------------------
// Kernel 1: partial scores[b,h,d,e] = sum_{s in chunk} K[b,h,d,s] * V[b,h,s,e]
// grid = (BSZ*NH, SPLIT), block = 32 (one wave). fp32 WMMA 16x16x4, 2x2 tiles.
// K/V are streamed exactly once -> non-temporal loads (keep L2 for weights).
// Strength-reduced addressing: per-lane base pointers + constant strides.
// ---------------------------------------------------------------------------
__global__ __launch_bounds__(32) void kv_gemm(const float* __restrict__ K,
                                              const float* __restrict__ V,
                                              float* __restrict__ partial) {
    const int bh     = blockIdx.x;      // 0..255
    const int chunk  = blockIdx.y;      // 0..SPLIT-1
    const int lane   = threadIdx.x;     // 0..31
    const int lane16 = lane & 15;
    const int laneHi = lane >> 4;       // 0 or 1

    const float* kp = K + (size_t)bh * HD * SEQ;  // (32, 4096) row-major
    const float* vp = V + (size_t)bh * SEQ * HD;  // (4096, 32) row-major

    const int s0 = chunk * (SEQ / SPLIT);

    // A layout (16x4 fp32): lanes 0-15 hold k=0,1 ; lanes 16-31 hold k=2,3.
    const float* aP0 = kp + (size_t)lane16 * SEQ + s0 + 2 * laneHi;        // d rows 0-15
    const float* aP1 = aP0 + (size_t)16 * SEQ;                             // d rows 16-31
    // B layout (4x16 fp32): VGPR0 row = sk, VGPR1 row = sk+1, col = lane16.
    const float* bP  = vp + (size_t)(s0 + 2 * laneHi) * HD + lane16;

    v8f acc00 = {}, acc01 = {}, acc10 = {}, acc11 = {};

    #pragma unroll 4
    for (int it = 0; it < (SEQ / SPLIT) / 4; ++it) {
        v2f a0 = __builtin_nontemporal_load((const v2f*)aP0);
        v2f a1 = __builtin_nontemporal_load((const v2f*)aP1);
        v2f b0, b1;
        b0.x = __builtin_nontemporal_load(bP);
        b0.y = __builtin_nontemporal_load(bP + HD);
        b1.x = __builtin_nontemporal_load(bP + 16);
        b1.y = __builtin_nontemporal_load(bP + HD + 16);

        acc00 = __builtin_amdgcn_wmma_f32_16x16x4_f32(false, a0, false, b0, (short)0, acc00, false, false);
        acc01 = __builtin_amdgcn_wmma_f32_16x16x4_f32(false, a0, false, b1, (short)0, acc01, false, false);
        acc10 = __builtin_amdgcn_wmma_f32_16x16x4_f32(false, a1, false, b0, (short)0, acc10, false, false);
        acc11 = __builtin_amdgcn_wmma_f32_16x16x4_f32(false, a1, false, b1, (short)0, acc11, false, false);

        aP0 += 4;            // advance 4 columns in K
        aP1 += 4;
        bP  += 4 * HD;       // advance 4 rows in V (512 B)
    }

    // Store C/D tiles: VGPR r holds row r (lanes 0-15) / row r+8 (lanes 16-31), col = lane16.
    float* sp = partial + (size_t)chunk * SCORE_N + (size_t)bh * HD * HD;
    #pragma unroll
    for (int r = 0; r < 8; ++r) {
        const int row = r + laneHi * 8;
        sp[(row +  0) * HD + lane16     ] = acc00[r];
        sp[(row +  0) * HD + lane16 + 16] = acc01[r];
        sp[(row + 16) * HD + lane16     ] = acc10[r];
        sp[(row + 16) * HD + lane16 + 16] = acc11[r];
    }
}

// ---------------------------------------------------------------------------
// Kernel 2: reduce SPLIT partials, scale by 1/sqrt(4096)=1/64, softmax over b
// (axis 0), write attn_weights to d_out tail. One thread per (h,d,e).
// ---------------------------------------------------------------------------
__global__ __launch_bounds__(256) void softmax_b(const float* __restrict__ partial,
                                                 float* __restrict__ wts) {
    const int idx = blockIdx.x * blockDim.x + threadIdx.x;  // 0 .. NH*HD*HD-1
    const float scale = 0.015625f;  // 1/64

    float x[BSZ];
    float m = -3.4e38f;
    #pragma unroll
    for (int b = 0; b < BSZ; ++b) {
        float s = 0.f;
        #pragma unroll
        for (int c = 0; c < SPLIT; ++c)
            s += partial[(size_t)c * SCORE_N + (size_t)b * (NH * HD * HD) + idx];
        x[b] = s * scale;
        m = fmaxf(m, x[b]);
    }
    float sum = 0.f;
    #pragma unroll
    for (int b = 0; b < BSZ; ++b) { x[b] = __expf(x[b] - m); sum += x[b]; }
    const float inv = 1.0f / sum;
    #pragma unroll
    for (int b = 0; b < BSZ; ++b)
        wts[(size_t)b * (NH * HD * HD) + idx] = x[b] * inv;
}

// ---------------------------------------------------------------------------
// Kernel 3: out[b,h,l,e] = sum_d Q[b,h,l,d] * W[b,h,d,e]
// grid = (BSZ*NH, SEQ/128), block = 256 (8 waves). Each wave: one 16-row tile,
// full N=32, K=32 in 8 WMMA steps; W (32x32) preloaded as B-tiles in registers.
// Q is streamed once (NT loads); Out written once (NT stores); W loads stay
// regular-temporal so the 1 MB weight image lives in the 192 MB L2 for all
// 65536 waves (8192 blocks).
// ---------------------------------------------------------------------------
__global__ __launch_bounds__(256) void q_gemm(const float* __restrict__ Q,
                                              const float* __restrict__ Wt,
                                              float* __restrict__ Out) {
    const int bh     = blockIdx.x;
    const int wave   = threadIdx.x >> 5;   // 0..7
    const int lane   = threadIdx.x & 31;
    const int lane16 = lane & 15;
    const int laneHi = lane >> 4;

    // Preload all 16 B-tiles of the 32x32 weight matrix (k-step x e-tile).
    const float* wp = Wt + (size_t)bh * HD * HD;
    v2f btile[8][2];
    #pragma unroll
    for (int kt = 0; kt < 8; ++kt) {
        const int krow = kt * 4 + 2 * laneHi;
        #pragma unroll
        for (int et = 0; et < 2; ++et) {
            btile[kt][et].x = wp[(krow + 0) * HD + et * 16 + lane16];
            btile[kt][et].y = wp[(krow + 1) * HD + et * 16 + lane16];
        }
    }

    const int m0 = (blockIdx.y * 8 + wave) * 16;  // row-tile base within SEQ
    const float* qp = Q + (size_t)bh * SEQ * HD + (size_t)m0 * HD
                        + (size_t)lane16 * HD + 2 * laneHi;

    v8f acc0 = {}, acc1 = {};
    #pragma unroll
    for (int kt = 0; kt < 8; ++kt) {
        // A tile: Q rows l (M), cols d (K); contiguous float2 per lane.
        v2f a = __builtin_nontemporal_load((const v2f*)(qp + kt * 4));
        acc0 = __builtin_amdgcn_wmma_f32_16x16x4_f32(false, a, false, btile[kt][0], (short)0, acc0, false, false);
        acc1 = __builtin_amdgcn_wmma_f32_16x16x4_f32(false, a, false, btile[kt][1], (short)0, acc1, false, false);
    }

    float* op = Out + (size_t)bh * SEQ * HD + (size_t)m0 * HD;
    #pragma unroll
    for (int r = 0; r < 8; ++r) {
        const int row = r + laneHi * 8;
        __builtin_nontemporal_store(acc0[r], op + row * HD + lane16);
        __builtin_nontemporal_store(acc1[r], op + row * HD + 16 + lane16);
    }
}

// ---------------------------------------------------------------------------
extern "C" void kernel_launch(void* const* d_in, const int* in_sizes, int n_in,
                              void* d_out, int out_size, void* d_ws, size_t ws_size,
                              hipStream_t stream) {
    const float* q = (const float*)d_in[0];  // (16,16,4096,32)
    const float* k = (const float*)d_in[1];  // (16,16,32,4096)
    const float* v = (const float*)d_in[2];  // (16,16,4096,32)

    float* out     = (float*)d_out;          // output  (16,16,4096,32)
    float* wts     = out + OUT_OFF;          // attn_weights (16,16,32,32)
    float* partial = (float*)d_ws;           // SPLIT * SCORE_N floats (8 MB)

    // 1) partial scores = K @ V (split over L, deterministic per-chunk stores)
    kv_gemm<<<dim3(BSZ * NH, SPLIT), 32, 0, stream>>>(k, v, partial);

    // 2) reduce + scale + softmax over batch axis -> attn_weights
    softmax_b<<<(NH * HD * HD) / 256, 256, 0, stream>>>(partial, wts);

    // 3) output = Q @ attn_weights
    q_gemm<<<dim3(BSZ * NH, SEQ / 128), 256, 0, stream>>>(q, wts, out);
}